// CRFDecoder_39505109188859
// MI455X (gfx1250) — compile-verified
//
#include <hip/hip_runtime.h>
#include <hip/hip_bf16.h>

typedef float v2f __attribute__((ext_vector_type(2)));
typedef float v8f __attribute__((ext_vector_type(8)));

#define BATCH      128
#define SEQ        512
#define DDIM       1024
#define NL         50          // LABEL_SIZE
#define NUM_LABELS 52
#define START_LBL  50
#define END_LBL    51
#define LSTRIDE    52          // padded logits row stride
#define ESTRIDE    53          // LDS stride for exp(T): avoids lane/lane+16 bank conflicts
#define NEGV       (-100.0f)
#define BS_TOTAL   (BATCH * SEQ)

// ---------------------------------------------------------------------------
// Kernel 1: logits = inputs @ W^T + b  via f32 WMMA (16x16x4).
// One wave computes a 16-row x 64-col tile: A fragment loaded once per k-step,
// reused by 4 WMMA accumulators. Block = 4 waves = 64 rows. Grid = 1024.
// ---------------------------------------------------------------------------
__global__ __launch_bounds__(128) void crf_gemm_logits(
    const float* __restrict__ A,     // (BS_TOTAL, DDIM)
    const float* __restrict__ W,     // (NL, DDIM)
    const float* __restrict__ bias,  // (NL)
    float* __restrict__ out)         // (BS_TOTAL, LSTRIDE), cols [0,NL) valid
{
  const int wave  = threadIdx.x >> 5;
  const int lane  = threadIdx.x & 31;
  const int nrow  = lane & 15;   // M row within A-tile / N col within B-tile
  const int khalf = lane >> 4;   // lane-half selects K {0,1} vs {2,3}
  const int m0    = (blockIdx.x * 4 + wave) * 16;

  const float* arow = A + (size_t)(m0 + nrow) * DDIM;
  const bool v3 = (nrow + 48) < NL;          // only cols 48,49 of tile 3 valid
  const float* w0p = W + (size_t)(nrow +  0) * DDIM;
  const float* w1p = W + (size_t)(nrow + 16) * DDIM;
  const float* w2p = W + (size_t)(nrow + 32) * DDIM;
  const float* w3p = W + (size_t)(v3 ? nrow + 48 : 0) * DDIM;  // clamp; junk cols never stored

  v8f acc0 = {}, acc1 = {}, acc2 = {}, acc3 = {};
  #pragma unroll 4
  for (int k = 0; k < DDIM; k += 4) {
    const int kk = k + khalf * 2;
    v2f a, b0, b1, b2, b3;
    // 32-bit 16x4 A layout: VGPR0 holds K = 0 / 2 (per lane half), VGPR1 holds K = 1 / 3
    a.x  = arow[kk]; a.y  = arow[kk + 1];
    b0.x = w0p[kk];  b0.y = w0p[kk + 1];
    b1.x = w1p[kk];  b1.y = w1p[kk + 1];
    b2.x = w2p[kk];  b2.y = w2p[kk + 1];
    b3.x = w3p[kk];  b3.y = w3p[kk + 1];
    acc0 = __builtin_amdgcn_wmma_f32_16x16x4_f32(false, a, false, b0, (short)0, acc0, false, false);
    acc1 = __builtin_amdgcn_wmma_f32_16x16x4_f32(false, a, false, b1, (short)0, acc1, false, false);
    acc2 = __builtin_amdgcn_wmma_f32_16x16x4_f32(false, a, false, b2, (short)0, acc2, false, false);
    acc3 = __builtin_amdgcn_wmma_f32_16x16x4_f32(false, a, false, b3, (short)0, acc3, false, false);
  }

  const float bb0 = bias[nrow];
  const float bb1 = bias[nrow + 16];
  const float bb2 = bias[nrow + 32];
  const float bb3 = v3 ? bias[nrow + 48] : 0.f;
  // C/D layout: VGPR r -> row m0 + r + 8*khalf, col = nrow (per lane group)
  #pragma unroll
  for (int r = 0; r < 8; ++r) {
    const int m = m0 + r + khalf * 8;
    float* orow = out + (size_t)m * LSTRIDE;
    orow[nrow +  0] = acc0[r] + bb0;
    orow[nrow + 16] = acc1[r] + bb1;
    orow[nrow + 32] = acc2[r] + bb2;
    if (v3) orow[nrow + 48] = acc3[r] + bb3;
  }
}

// ---------------------------------------------------------------------------
// Kernel 2: per-batch CRF forward scan + gold score. 1 block (64 thr) / batch.
// logsumexp_j(alpha_j + T_ij) == m + log( sum_j exp(T_ij) * exp(alpha_j - m) )
// with E = exp(T) precomputed in LDS -> the serial step is a 52x52 FMA mat-vec.
// ---------------------------------------------------------------------------
__global__ __launch_bounds__(64) void crf_scan(
    const float* __restrict__ logits,   // (BS_TOTAL, LSTRIDE)
    const float* __restrict__ T,        // (NUM_LABELS, NUM_LABELS)
    const int*   __restrict__ lens,     // (BATCH)
    const int*   __restrict__ labels,   // (BATCH, SEQ)
    float*       __restrict__ partial)  // (BATCH) : gold - norm
{
  __shared__ float E[NUM_LABELS * ESTRIDE];
  __shared__ float alpha[NUM_LABELS];
  __shared__ float p[NUM_LABELS];
  __shared__ float red[8];

  const int b   = blockIdx.x;
  const int tid = threadIdx.x;

  for (int idx = tid; idx < NUM_LABELS * NUM_LABELS; idx += 64) {
    const int i = idx / NUM_LABELS, j = idx % NUM_LABELS;
    E[i * ESTRIDE + j] = __expf(T[idx]);     // T=-100 underflows to 0: correct
  }
  if (tid < NUM_LABELS) alpha[tid] = (tid == START_LBL) ? 0.f : NEGV;
  __syncthreads();

  const int len = lens[b];
  const int* lab = labels + (size_t)b * SEQ;
  const float* lrow = logits + (size_t)b * SEQ * LSTRIDE;

  for (int t = 0; t < len; ++t) {   // steps with t >= len leave alpha unchanged
    const float a_i = (tid < NUM_LABELS) ? alpha[tid] : NEGV;
    // global max of alpha across both waves
    float mx = a_i;
    #pragma unroll
    for (int off = 16; off; off >>= 1) mx = fmaxf(mx, __shfl_xor(mx, off, 32));
    if ((tid & 31) == 0) red[tid >> 5] = mx;
    __syncthreads();
    const float m = fmaxf(red[0], red[1]);
    if (tid < NUM_LABELS) p[tid] = __expf(a_i - m);
    __syncthreads();

    float alpha_new = NEGV;
    if (tid < NUM_LABELS) {
      const float* Er = E + tid * ESTRIDE;
      float acc = 0.f;
      #pragma unroll 4
      for (int j = 0; j < NUM_LABELS; ++j) acc = fmaf(Er[j], p[j], acc);
      const float lg = (tid < NL) ? lrow[(size_t)t * LSTRIDE + tid] : NEGV;
      alpha_new = m + __logf(fmaxf(acc, 1e-37f)) + lg;
    }
    __syncthreads();
    if (tid < NUM_LABELS) alpha[tid] = alpha_new;
    __syncthreads();
  }

  // norm = logsumexp(alpha + T[END, :])
  const float v = (tid < NUM_LABELS) ? alpha[tid] + T[END_LBL * NUM_LABELS + tid] : NEGV;
  float mx = v;
  #pragma unroll
  for (int off = 16; off; off >>= 1) mx = fmaxf(mx, __shfl_xor(mx, off, 32));
  if ((tid & 31) == 0) red[tid >> 5] = mx;
  __syncthreads();
  const float m = fmaxf(red[0], red[1]);
  float e = (tid < NUM_LABELS) ? __expf(v - m) : 0.f;
  #pragma unroll
  for (int off = 16; off; off >>= 1) e += __shfl_xor(e, off, 32);
  if ((tid & 31) == 0) red[2 + (tid >> 5)] = e;
  __syncthreads();
  const float norm = m + __logf(red[2] + red[3]);

  // gold score: unary gathers + transition pairs (parallel over 64 threads)
  float acc = 0.f;
  for (int k = tid; k < len; k += 64) acc += lrow[(size_t)k * LSTRIDE + lab[k]];
  for (int k = tid; k <= len; k += 64) {
    const int prev = (k == 0) ? START_LBL : lab[k - 1];
    const int next = (k < len) ? lab[k] : END_LBL;
    acc += T[next * NUM_LABELS + prev];
  }
  #pragma unroll
  for (int off = 16; off; off >>= 1) acc += __shfl_xor(acc, off, 32);
  if ((tid & 31) == 0) red[4 + (tid >> 5)] = acc;
  __syncthreads();
  if (tid == 0) partial[b] = (red[4] + red[5]) - norm;
}

// ---------------------------------------------------------------------------
// Kernel 3: loss = -sum_b partial[b]
// ---------------------------------------------------------------------------
__global__ __launch_bounds__(128) void crf_final_reduce(
    const float* __restrict__ partial, float* __restrict__ out)
{
  __shared__ float s[4];
  const int tid = threadIdx.x;
  float v = partial[tid];
  #pragma unroll
  for (int off = 16; off; off >>= 1) v += __shfl_xor(v, off, 32);
  if ((tid & 31) == 0) s[tid >> 5] = v;
  __syncthreads();
  if (tid == 0) out[0] = -(s[0] + s[1] + s[2] + s[3]);
}

// ---------------------------------------------------------------------------
extern "C" void kernel_launch(void* const* d_in, const int* in_sizes, int n_in,
                              void* d_out, int out_size, void* d_ws, size_t ws_size,
                              hipStream_t stream) {
  const float* inputs = (const float*)d_in[0];   // (B,S,D)
  const float* W      = (const float*)d_in[1];   // (NL,D)
  const float* bias   = (const float*)d_in[2];   // (NL)
  const float* T      = (const float*)d_in[3];   // (52,52)
  const int*   lens   = (const int*)d_in[4];     // (B)
  const int*   labels = (const int*)d_in[5];     // (B,S)

  float* logits_ws = (float*)d_ws;                           // BS_TOTAL * LSTRIDE floats
  float* partial   = logits_ws + (size_t)BS_TOTAL * LSTRIDE; // BATCH floats

  crf_gemm_logits<<<BS_TOTAL / 64, 128, 0, stream>>>(inputs, W, bias, logits_ws);
  crf_scan<<<BATCH, 64, 0, stream>>>(logits_ws, T, lens, labels, partial);
  crf_final_reduce<<<1, BATCH, 0, stream>>>(partial, (float*)d_out);
}